// GAT_652835029007
// MI455X (gfx1250) — compile-verified
//
#include <hip/hip_runtime.h>
#include <cmath>

#define NN   20001     // nodes (+ padding row)
#define INF_ 256       // in_features
#define HIDDEN 128     // per-head hidden
#define NH   4         // heads
#define NB   2         // batch
#define NE   500000    // edges per graph
#define FEAT 512       // NH*HIDDEN
#define INV_SCALE 0.04419417382415922f   // 1/sqrt(512)
#define LN_EPS 1e-6f

typedef float v2f __attribute__((ext_vector_type(2)));
typedef float v8f __attribute__((ext_vector_type(8)));

// ---------------------------------------------------------------------------
// GEMM via V_WMMA_F32_16X16X4_F32: C[M,NOUT] = A[M,K] @ Wt[NOUT,K]^T + bias
// One wave computes a 16(M) x 64(N) tile; K stepped by 4, 1-stage pipelined.
// Out-of-range M rows are clamped (their D rows are never stored), so all
// fragment loads are unpredicated b64 loads -> no exec-mask divergence.
// ---------------------------------------------------------------------------
__global__ void __launch_bounds__(256) GAT_gemm_wmma(
    const float* __restrict__ A, const float* __restrict__ Wt,
    const float* __restrict__ bias, float* __restrict__ C,
    int M, int K, int NOUT)
{
  const int wavesPerBlock = blockDim.x >> 5;
  int wid = blockIdx.x * wavesPerBlock + (threadIdx.x >> 5);
  int nTiles = NOUT >> 6;
  int mTile = wid / nTiles;
  int nTile = wid - mTile * nTiles;
  int row0 = mTile * 16;
  if (row0 >= M) return;                      // wave-uniform exit (EXEC stays all-1)
  int L    = threadIdx.x & 31;
  int mL   = L & 15;
  int kSel = (L >> 4) << 1;                   // lanes 0-15 -> k{0,1}; 16-31 -> k{2,3}
  int row  = row0 + mL;
  if (row >= M) row = M - 1;                  // clamp: keeps loads in-bounds, D row unused
  const float* ap  = A + (size_t)row * K + kSel;
  int col0 = nTile << 6;
  const float* bp0 = Wt + (size_t)(col0 + mL) * K + kSel;
  const float* bp1 = bp0 + (size_t)16 * K;
  const float* bp2 = bp0 + (size_t)32 * K;
  const float* bp3 = bp0 + (size_t)48 * K;

  v8f acc0 = {}, acc1 = {}, acc2 = {}, acc3 = {};

  v2f a, b0, b1, b2, b3;
  a.x  = ap[0];  a.y  = ap[1];
  b0.x = bp0[0]; b0.y = bp0[1];
  b1.x = bp1[0]; b1.y = bp1[1];
  b2.x = bp2[0]; b2.y = bp2[1];
  b3.x = bp3[0]; b3.y = bp3[1];

  for (int k = 4; k < K; k += 4) {
    v2f an, b0n, b1n, b2n, b3n;               // prefetch next k-step
    an.x  = ap[k];  an.y  = ap[k + 1];
    b0n.x = bp0[k]; b0n.y = bp0[k + 1];
    b1n.x = bp1[k]; b1n.y = bp1[k + 1];
    b2n.x = bp2[k]; b2n.y = bp2[k + 1];
    b3n.x = bp3[k]; b3n.y = bp3[k + 1];
    acc0 = __builtin_amdgcn_wmma_f32_16x16x4_f32(false, a, false, b0, (short)0, acc0, false, false);
    acc1 = __builtin_amdgcn_wmma_f32_16x16x4_f32(false, a, false, b1, (short)0, acc1, false, false);
    acc2 = __builtin_amdgcn_wmma_f32_16x16x4_f32(false, a, false, b2, (short)0, acc2, false, false);
    acc3 = __builtin_amdgcn_wmma_f32_16x16x4_f32(false, a, false, b3, (short)0, acc3, false, false);
    a = an; b0 = b0n; b1 = b1n; b2 = b2n; b3 = b3n;
  }
  acc0 = __builtin_amdgcn_wmma_f32_16x16x4_f32(false, a, false, b0, (short)0, acc0, false, false);
  acc1 = __builtin_amdgcn_wmma_f32_16x16x4_f32(false, a, false, b1, (short)0, acc1, false, false);
  acc2 = __builtin_amdgcn_wmma_f32_16x16x4_f32(false, a, false, b2, (short)0, acc2, false, false);
  acc3 = __builtin_amdgcn_wmma_f32_16x16x4_f32(false, a, false, b3, (short)0, acc3, false, false);

  int mAdd = (L < 16) ? 0 : 8;
  #pragma unroll
  for (int r = 0; r < 8; ++r) {
    int rr = row0 + r + mAdd;
    if (rr < M) {
      size_t base = (size_t)rr * NOUT + col0 + mL;
      C[base]      = acc0[r] + bias[col0 +      mL];
      C[base + 16] = acc1[r] + bias[col0 + 16 + mL];
      C[base + 32] = acc2[r] + bias[col0 + 32 + mL];
      C[base + 48] = acc3[r] + bias[col0 + 48 + mL];
    }
  }
}

// ---------------------------------------------------------------------------
// Per-(head,node) attention dots:  osrc[h,v]=x.a_src  odst[h,v]=x.a_dst
// ---------------------------------------------------------------------------
__global__ void GAT_dots(const float* __restrict__ X, const float* __restrict__ a,
                         float* __restrict__ osrc, float* __restrict__ odst)
{
  int idx = blockIdx.x * blockDim.x + threadIdx.x;
  if (idx >= NH * NN) return;
  int h = idx / NN, v = idx - h * NN;
  const float* xr = X + (size_t)v * FEAT + h * HIDDEN;
  const float* as = a + h * 2 * HIDDEN;
  const float* ad = as + HIDDEN;
  float s1 = 0.0f, s2 = 0.0f;
  for (int c = 0; c < HIDDEN; ++c) { float xv = xr[c]; s1 += xv * as[c]; s2 += xv * ad[c]; }
  osrc[idx] = s1; odst[idx] = s2;
}

__global__ void GAT_zero(int* __restrict__ deg, float* __restrict__ hp2, float* __restrict__ rs2)
{
  int i = blockIdx.x * blockDim.x + threadIdx.x;
  if (i < NB * NN)   deg[i] = 0;
  if (i < NB * FEAT) hp2[i] = 0.0f;
  if (i < NB * NH)   rs2[i] = 0.0f;
}

__global__ void GAT_count(const int* __restrict__ edges, int* __restrict__ deg)
{
  int i = blockIdx.x * blockDim.x + threadIdx.x;
  if (i >= NB * NE) return;
  int b = i / NE, e = i - b * NE;
  int src = edges[(size_t)b * 2 * NE + e];
  atomicAdd(&deg[b * NN + src], 1);
}

// one block per graph: exclusive scan of degrees -> offsets + scatter cursors
__global__ void __launch_bounds__(1024) GAT_scan(const int* __restrict__ deg,
                                                 int* __restrict__ offs, int* __restrict__ curs)
{
  int b = blockIdx.x;
  __shared__ int sd[1024];
  __shared__ int carry;
  int tid = threadIdx.x;
  if (tid == 0) carry = 0;
  __syncthreads();
  for (int base = 0; base < NN; base += 1024) {
    int i = base + tid;
    int v = (i < NN) ? deg[b * NN + i] : 0;
    sd[tid] = v;
    __syncthreads();
    for (int off = 1; off < 1024; off <<= 1) {
      int t = (tid >= off) ? sd[tid - off] : 0;
      __syncthreads();
      sd[tid] += t;
      __syncthreads();
    }
    int incl = sd[tid];
    int c0 = carry;
    if (i < NN) { int ex = c0 + incl - v; offs[b * (NN + 1) + i] = ex; curs[b * NN + i] = ex; }
    __syncthreads();
    if (tid == 0) carry = c0 + sd[1023];
    __syncthreads();
  }
  if (tid == 0) offs[b * (NN + 1) + NN] = carry;
}

// per-edge: compute exp(leakyrelu(.)/scale) for all heads, scatter into CSR slot
__global__ void GAT_edge_scatter(const int* __restrict__ edges,
                                 const float* __restrict__ sasrc, const float* __restrict__ sadst,
                                 int* __restrict__ curs, int* __restrict__ dsts,
                                 float4* __restrict__ es)
{
  int i = blockIdx.x * blockDim.x + threadIdx.x;
  if (i >= NB * NE) return;
  int b = i / NE, e = i - b * NE;
  const int* eb = edges + (size_t)b * 2 * NE;
  int src = eb[e];
  int dst = eb[NE + e];
  float ev[NH];
  #pragma unroll
  for (int h = 0; h < NH; ++h) {
    float s = sasrc[h * NN + src] + sadst[h * NN + dst];
    float l = (s > 0.0f) ? s : 0.2f * s;
    ev[h] = expf(l * INV_SCALE);
  }
  int pos = atomicAdd(&curs[b * NN + src], 1);
  dsts[(size_t)b * NE + pos] = dst;
  es[(size_t)b * NE + pos] = make_float4(ev[0], ev[1], ev[2], ev[3]);
}

// one wave per (graph,node): hp[v,:] = sum_e e*x1[dst] / sum_e e.
// Atomic-free; edge loop software-pipelined one stage so the 2KB row gather
// for edge j+1 is in flight while edge j's FMAs retire. All pipeline guards
// are wave-uniform (CSR range is per-wave) -> no exec divergence.
__global__ void GAT_agg1(const int* __restrict__ offs, const int* __restrict__ dsts,
                         const float* __restrict__ es, const float* __restrict__ X1,
                         float* __restrict__ hp)
{
  int wid = (blockIdx.x * blockDim.x + threadIdx.x) >> 5;
  if (wid >= NB * NN) return;
  int b = wid / NN, v = wid - b * NN;
  int L = threadIdx.x & 31;
  int head = L >> 3;                       // lane owns cols [16L,16L+16) -> head = L/8
  int beg = offs[b * (NN + 1) + v];
  int end = offs[b * (NN + 1) + v + 1];
  const int*   db = dsts + (size_t)b * NE;
  const float* eb = es   + (size_t)b * NE * 4;
  float acc[16];
  #pragma unroll
  for (int k = 0; k < 16; ++k) acc[k] = 0.0f;
  float rs = 0.0f;
  if (beg < end) {
    int d = db[beg];
    float e = eb[(size_t)beg * 4 + head];
    const float4* xr = (const float4*)(X1 + (size_t)d * FEAT + L * 16);
    float4 r0 = xr[0], r1 = xr[1], r2 = xr[2], r3 = xr[3];
    for (int j = beg + 1; j <= end; ++j) {
      bool more = (j < end);               // wave-uniform
      int jn = more ? j : beg;             // tail: re-read first slot (in-bounds, L2-hot)
      int dn = db[jn];
      float en = more ? eb[(size_t)j * 4 + head] : 0.0f;
      const float4* xn = (const float4*)(X1 + (size_t)dn * FEAT + L * 16);
      float4 n0 = xn[0], n1 = xn[1], n2 = xn[2], n3 = xn[3];
      acc[0]  += e * r0.x; acc[1]  += e * r0.y; acc[2]  += e * r0.z; acc[3]  += e * r0.w;
      acc[4]  += e * r1.x; acc[5]  += e * r1.y; acc[6]  += e * r1.z; acc[7]  += e * r1.w;
      acc[8]  += e * r2.x; acc[9]  += e * r2.y; acc[10] += e * r2.z; acc[11] += e * r2.w;
      acc[12] += e * r3.x; acc[13] += e * r3.y; acc[14] += e * r3.z; acc[15] += e * r3.w;
      rs += e;
      r0 = n0; r1 = n1; r2 = n2; r3 = n3; e = en;
    }
  }
  if (rs == 0.0f) rs = 1.0f;
  float4* outp = (float4*)(hp + (size_t)wid * FEAT + L * 16);
  #pragma unroll
  for (int q = 0; q < 4; ++q)
    outp[q] = make_float4(acc[4*q+0]/rs, acc[4*q+1]/rs, acc[4*q+2]/rs, acc[4*q+3]/rs);
}

// in-place LayerNorm (unbiased std, eps on std) + ELU, one wave per row
__global__ void GAT_ln_elu(float* __restrict__ hc, const float* __restrict__ g,
                           const float* __restrict__ bb)
{
  int wid = (blockIdx.x * blockDim.x + threadIdx.x) >> 5;
  if (wid >= NB * NN) return;
  int L = threadIdx.x & 31;
  float4* rp = (float4*)(hc + (size_t)wid * FEAT + L * 16);
  float x[16];
  #pragma unroll
  for (int q = 0; q < 4; ++q) {
    float4 p = rp[q];
    x[4*q+0]=p.x; x[4*q+1]=p.y; x[4*q+2]=p.z; x[4*q+3]=p.w;
  }
  float s = 0.0f;
  #pragma unroll
  for (int k = 0; k < 16; ++k) s += x[k];
  #pragma unroll
  for (int m = 16; m >= 1; m >>= 1) s += __shfl_xor(s, m, 32);
  float mean = s * (1.0f / FEAT);
  float vs = 0.0f;
  #pragma unroll
  for (int k = 0; k < 16; ++k) { float d = x[k] - mean; vs += d * d; }
  #pragma unroll
  for (int m = 16; m >= 1; m >>= 1) vs += __shfl_xor(vs, m, 32);
  float inv = 1.0f / (sqrtf(vs * (1.0f / (FEAT - 1))) + LN_EPS);
  #pragma unroll
  for (int k = 0; k < 16; ++k) {
    int col = L * 16 + k;
    float v = g[col] * (x[k] - mean) * inv + bb[col];
    x[k] = (v > 0.0f) ? v : (expf(v) - 1.0f);
  }
  #pragma unroll
  for (int q = 0; q < 4; ++q)
    rp[q] = make_float4(x[4*q+0], x[4*q+1], x[4*q+2], x[4*q+3]);
}

// layer-2 aggregation: only src==N-1 contributes to the output row
__global__ void GAT_agg2_last(const int* __restrict__ edges,
                              const float* __restrict__ sasrc, const float* __restrict__ sadst,
                              const float* __restrict__ X2,
                              float* __restrict__ hp2, float* __restrict__ rs2)
{
  int i = blockIdx.x * blockDim.x + threadIdx.x;
  if (i >= NB * NE) return;
  int b = i / NE, e = i - b * NE;
  const int* eb = edges + (size_t)b * 2 * NE;
  int src = eb[e];
  if (src != NN - 1) return;
  int dst = eb[NE + e];
  const float* sas = sasrc + (size_t)b * NH * NN;
  const float* sad = sadst + (size_t)b * NH * NN;
  const float* xr  = X2 + ((size_t)b * NN + dst) * FEAT;
  #pragma unroll
  for (int h = 0; h < NH; ++h) {
    float s = sas[h * NN + (NN - 1)] + sad[h * NN + dst];
    float l = (s > 0.0f) ? s : 0.2f * s;
    float ev = expf(l * INV_SCALE);
    atomicAdd(&rs2[b * NH + h], ev);
    for (int c = 0; c < HIDDEN; ++c)
      atomicAdd(&hp2[b * FEAT + h * HIDDEN + c], ev * xr[h * HIDDEN + c]);
  }
}

// mean over heads -> LN -> relu -> 2-class projection; one 128-thread block/graph
__global__ void GAT_final(const float* __restrict__ hp2, const float* __restrict__ rs2,
                          const float* __restrict__ g2, const float* __restrict__ bn2,
                          const float* __restrict__ Vw, const float* __restrict__ Vb,
                          float* __restrict__ out)
{
  int b = blockIdx.x;
  int c = threadIdx.x;                       // 0..127
  __shared__ float red[128];
  float m2 = 0.0f;
  #pragma unroll
  for (int h = 0; h < NH; ++h) {
    float r = rs2[b * NH + h];
    if (r == 0.0f) r = 1.0f;
    m2 += hp2[b * FEAT + h * HIDDEN + c] / r;
  }
  m2 *= (1.0f / NH);
  red[c] = m2; __syncthreads();
  for (int o = 64; o > 0; o >>= 1) { if (c < o) red[c] += red[c + o]; __syncthreads(); }
  float mean = red[0] * (1.0f / HIDDEN);
  __syncthreads();
  float d = m2 - mean;
  red[c] = d * d; __syncthreads();
  for (int o = 64; o > 0; o >>= 1) { if (c < o) red[c] += red[c + o]; __syncthreads(); }
  float stdv = sqrtf(red[0] * (1.0f / (HIDDEN - 1)));
  __syncthreads();
  float ln = g2[c] * (m2 - mean) / (stdv + LN_EPS) + bn2[c];
  float r = ln > 0.0f ? ln : 0.0f;
  red[c] = r * Vw[c]; __syncthreads();
  for (int o = 64; o > 0; o >>= 1) { if (c < o) red[c] += red[c + o]; __syncthreads(); }
  if (c == 0) out[b * 2 + 0] = red[0] + Vb[0];
  __syncthreads();
  red[c] = r * Vw[HIDDEN + c]; __syncthreads();
  for (int o = 64; o > 0; o >>= 1) { if (c < o) red[c] += red[c + o]; __syncthreads(); }
  if (c == 0) out[b * 2 + 1] = red[0] + Vb[1];
}

// ---------------------------------------------------------------------------
extern "C" void kernel_launch(void* const* d_in, const int* in_sizes, int n_in,
                              void* d_out, int out_size, void* d_ws, size_t ws_size,
                              hipStream_t stream)
{
  (void)in_sizes; (void)n_in; (void)out_size; (void)ws_size;
  const int*   edge_in  = (const int*)  d_in[0];
  const int*   edge_out = (const int*)  d_in[1];
  const float* embed    = (const float*)d_in[2];
  const float* W1  = (const float*)d_in[3];
  const float* b1  = (const float*)d_in[4];
  const float* a1  = (const float*)d_in[5];
  const float* g1  = (const float*)d_in[6];
  const float* bn1 = (const float*)d_in[7];
  const float* W2  = (const float*)d_in[8];
  const float* b2  = (const float*)d_in[9];
  const float* a2  = (const float*)d_in[10];
  const float* g2  = (const float*)d_in[11];
  const float* bn2 = (const float*)d_in[12];
  const float* Vw  = (const float*)d_in[13];
  const float* Vb  = (const float*)d_in[14];
  float* out = (float*)d_out;

  char* ws = (char*)d_ws;
  size_t off = 0;
  auto alloc = [&](size_t nElem) -> char* {
    char* p = ws + off;
    off += ((nElem * 4 + 255) & ~(size_t)255);
    return p;
  };
  float* X1   = (float*)alloc((size_t)NN * FEAT);        // embed @ W1^T (batch-invariant)
  float* SA1S = (float*)alloc((size_t)NH * NN);
  float* SA1D = (float*)alloc((size_t)NH * NN);
  float* HCAT = (float*)alloc((size_t)NB * NN * FEAT);   // hp1 -> (in place) LN+ELU
  float* X2   = (float*)alloc((size_t)NB * NN * FEAT);
  float* SA2S = (float*)alloc((size_t)NB * NH * NN);
  float* SA2D = (float*)alloc((size_t)NB * NH * NN);
  float* HP2  = (float*)alloc((size_t)NB * FEAT);
  float* RS2  = (float*)alloc((size_t)NB * NH);
  int*   DEG  = (int*)  alloc((size_t)NB * NN);
  int*   OFFS = (int*)  alloc((size_t)NB * (NN + 1));
  int*   CURS = (int*)  alloc((size_t)NB * NN);
  int*   DSTS = (int*)  alloc((size_t)NB * NE);
  float* ES   = (float*)alloc((size_t)NB * NE * 4);

  const int mtiles     = (NN + 15) / 16;
  const int gemmBlocks = (mtiles * (FEAT / 64) * 32 + 255) / 256;
  const int edgeBlocks = (NB * NE + 255) / 256;
  const int waveBlocks = (NB * NN * 32 + 255) / 256;

  GAT_zero<<<(NB * NN + 255) / 256, 256, 0, stream>>>(DEG, HP2, RS2);
  // layer 1 dense: x1 = embed @ W1^T + b1 (heads concatenated along out dim)
  GAT_gemm_wmma<<<gemmBlocks, 256, 0, stream>>>(embed, W1, b1, X1, NN, INF_, FEAT);
  GAT_dots<<<(NH * NN + 255) / 256, 256, 0, stream>>>(X1, a1, SA1S, SA1D);
  // CSR build on edge_in
  GAT_count<<<edgeBlocks, 256, 0, stream>>>(edge_in, DEG);
  GAT_scan<<<NB, 1024, 0, stream>>>(DEG, OFFS, CURS);
  GAT_edge_scatter<<<edgeBlocks, 256, 0, stream>>>(edge_in, SA1S, SA1D, CURS, DSTS, (float4*)ES);
  // atomic-free per-node aggregation + rowsum normalization
  GAT_agg1<<<waveBlocks, 256, 0, stream>>>(OFFS, DSTS, ES, X1, HCAT);
  GAT_ln_elu<<<waveBlocks, 256, 0, stream>>>(HCAT, g1, bn1);
  // layer 2 dense per graph: x2 = hcat @ W2^T + b2
  for (int b = 0; b < NB; ++b) {
    GAT_gemm_wmma<<<gemmBlocks, 256, 0, stream>>>(HCAT + (size_t)b * NN * FEAT, W2, b2,
                                                  X2 + (size_t)b * NN * FEAT, NN, FEAT, FEAT);
    GAT_dots<<<(NH * NN + 255) / 256, 256, 0, stream>>>(X2 + (size_t)b * NN * FEAT, a2,
                                                        SA2S + (size_t)b * NH * NN,
                                                        SA2D + (size_t)b * NH * NN);
  }
  // only src == N-1 reaches the output row
  GAT_agg2_last<<<edgeBlocks, 256, 0, stream>>>(edge_out, SA2S, SA2D, X2, HP2, RS2);
  GAT_final<<<NB, 128, 0, stream>>>(HP2, RS2, g2, bn2, Vw, Vb, out);
}